// CorrelationLayer_15324443312758
// MI455X (gfx1250) — compile-verified
//
#include <hip/hip_runtime.h>
#include <hip/hip_bf16.h>

typedef __attribute__((ext_vector_type(2))) float v2f;
typedef __attribute__((ext_vector_type(4))) float v4f;
typedef __attribute__((ext_vector_type(8))) float v8f;
typedef __attribute__((ext_vector_type(4))) int   v4i;

typedef __attribute__((address_space(1))) v4i* gbl4;   // global int4*
typedef __attribute__((address_space(3))) v4i* lds4;   // LDS int4*

constexpr int kB   = 8;
constexpr int kC   = 128;
constexpr int kH   = 192;
constexpr int kW   = 256;
constexpr int kD   = 40;
constexpr int kNC  = 41;     // D+1 displacements
constexpr int WBLK = 64;     // w columns per workgroup
constexpr int KCH  = 32;     // channel chunk staged in LDS (double buffered)
constexpr int NCHK = kC / KCH;
constexpr int ASTR = 72;     // LDS a stride (dwords): 2*72 % 64 == 16 -> disjoint bank halves
constexpr int BCOL = 112;    // LDS b cols/stride:     2*112 % 64 == 32 -> disjoint bank halves
constexpr int XSTR = 33;     // extraction buffer stride (34m mod 64 all distinct)

// ---- CDNA5 async copy: global -> LDS, 16B per lane, tracked by ASYNCcnt ----
__device__ __forceinline__ void async_copy16(const float* g, float* l) {
#if __has_builtin(__builtin_amdgcn_global_load_async_to_lds_b128)
  __builtin_amdgcn_global_load_async_to_lds_b128((gbl4)g, (lds4)l, 0, 0);
#else
  asm volatile("global_load_async_to_lds_b128 %0, %1, off"
               :: "v"((unsigned)(unsigned long long)(lds4)l),
                  "v"((unsigned long long)g)
               : "memory");
#endif
}

#if __has_builtin(__builtin_amdgcn_s_wait_asynccnt)
#define WAIT_ASYNC(n) __builtin_amdgcn_s_wait_asynccnt(n)
#else
#define WAIT_ASYNC(n) asm volatile("s_wait_asynccnt %0" :: "n"(n) : "memory")
#endif

__global__ __launch_bounds__(128, 1)
void corr_wmma_f32(const float* __restrict__ A, const float* __restrict__ Bm,
                   float* __restrict__ Out) {
  __shared__ float ash[2][KCH * ASTR];     // 18,432 B (double buffered)
  __shared__ float bsh[2][KCH * BCOL];     // 28,672 B (double buffered)
  __shared__ float xbuf[4 * 16 * XSTR];    //  8,448 B (per-wave regions)
  __shared__ float dump[128];              //    512 B (async OOB dump)

  const int tid  = threadIdx.x;
  const int lane = tid & 31;
  const int wv   = tid >> 5;        // wave id 0..3 -> w-subtile
  const int half = lane >> 4;       // K-row half select in WMMA frags
  const int nl   = lane & 15;

  const int w0 = blockIdx.x * WBLK;
  const int h  = blockIdx.y;
  const int bb = blockIdx.z;

  const int HW   = kH * kW;
  const int base = bb * (kC * HW) + h * kW;   // + c*HW + w

  // Pre-zero b buffers (pad columns must read 0; async loads never touch
  // OOB slots because those lanes are redirected to `dump`).
  for (int i = tid; i < 2 * KCH * BCOL; i += 128) (&bsh[0][0])[i] = 0.0f;
  __syncthreads();   // ds-store fence: zeros land before any async write

  // Issue one chunk's async copies: ALWAYS 11 instructions per thread
  // (uniform per wave -> exact ASYNCcnt bookkeeping).
  auto issue_chunk = [&](int kc, int sel) {
    const int cbase = kc * KCH;
    #pragma unroll
    for (int it = 0; it < 4; ++it) {            // a: KCH x 64 floats
      const int i  = tid + it * 128;
      const int c  = i >> 4;
      const int x4 = (i & 15) << 2;
      async_copy16(A + base + (cbase + c) * HW + w0 + x4,
                   &ash[sel][c * ASTR + x4]);
    }
    #pragma unroll
    for (int it = 0; it < 7; ++it) {            // b: KCH x 112 cols @ (w0-40)
      const int i   = tid + it * 128;
      const int c   = i / 28;
      const int x4  = (i - c * 28) << 2;
      const int col = w0 - kD + x4;
      const bool ok = (col >= 0) && (col + 3 < kW);
      const float* g = ok ? (Bm + base + (cbase + c) * HW + col) : Bm;
      float*       l = ok ? &bsh[sel][c * BCOL + x4] : &dump[lane * 4];
      async_copy16(g, l);
    }
  };

  v8f acc[3][2];
  #pragma unroll
  for (int ct = 0; ct < 3; ++ct) { acc[ct][0] = {}; acc[ct][1] = {}; }

  issue_chunk(0, 0);

  for (int kc = 0; kc < NCHK; ++kc) {
    const int sel = kc & 1;
    if (kc + 1 < NCHK) {
      issue_chunk(kc + 1, sel ^ 1);   // prefetch next chunk into other buffer
      WAIT_ASYNC(11);                 // in-order done -> chunk kc fully landed
    } else {
      WAIT_ASYNC(0);
    }
    __syncthreads();                  // every wave's chunk kc is visible

    // Gram-band accumulation: wave wv owns w-subtile [w0+16*wv, +16).
    const float* ab = ash[sel];
    const float* bbuf = bsh[sel];
    const int aoff = wv * 16 + nl;
    #pragma unroll
    for (int ct = 0; ct < 3; ++ct) {
      const int bcol0 = wv * 16 + ct * 16 + nl;   // LDS col of N=0
      v8f c0 = acc[ct][0];
      v8f c1 = acc[ct][1];
      #pragma unroll
      for (int k = 0; k < KCH; k += 4) {
        const int ka = k + 2 * half;   // 16x4 f32 layout: v0=K{0,2}, v1=K{1,3}
        v2f af, b0, b1;
        af.x = ab[ka * ASTR + aoff];
        af.y = ab[(ka + 1) * ASTR + aoff];
        b0.x = bbuf[ka * BCOL + bcol0];
        b0.y = bbuf[(ka + 1) * BCOL + bcol0];
        b1.x = bbuf[ka * BCOL + bcol0 + 16];
        b1.y = bbuf[(ka + 1) * BCOL + bcol0 + 16];
        c0 = __builtin_amdgcn_wmma_f32_16x16x4_f32(false, af, false, b0,
                                                   (short)0, c0, false, false);
        c1 = __builtin_amdgcn_wmma_f32_16x16x4_f32(false, af, false, b1,
                                                   (short)0, c1, false, false);
      }
      acc[ct][0] = c0;
      acc[ct][1] = c1;
    }
    __syncthreads();   // all waves done reading buffer sel before it is refilled
  }

  // Diagonal extraction via per-wave LDS buffer -> coalesced stores.
  // DS ops from one wave execute in order, so no barrier is required here.
  float* xb = &xbuf[wv * 16 * XSTR];
  const int wbase = w0 + wv * 16;
  #pragma unroll
  for (int ct = 0; ct < 3; ++ct) {
    const int ctr0 = ct * 16;
    #pragma unroll
    for (int r = 0; r < 8; ++r) {
      const int M = r + 8 * half;        // C/D layout: VGPR r holds rows r, r+8
      xb[M * XSTR + nl]      = acc[ct][0][r];   // N = nl
      xb[M * XSTR + 16 + nl] = acc[ct][1][r];   // N = 16 + nl
    }
    #pragma unroll
    for (int t = 0; t < 8; ++t) {
      const int idx = t * 32 + lane;
      const int j   = idx >> 4;          // displacement offset within tile
      const int m   = idx & 15;          // w offset within subtile
      const int ctr = ctr0 + j;
      if (ctr < kNC) {
        // out[w,ctr] = G[m][m+j]; each half-wave writes 16 contiguous dwords.
        Out[((bb * kNC + ctr) * kH + h) * kW + wbase + m] =
            xb[m * XSTR + m + j];
      }
    }
  }
}

extern "C" void kernel_launch(void* const* d_in, const int* in_sizes, int n_in,
                              void* d_out, int out_size, void* d_ws, size_t ws_size,
                              hipStream_t stream) {
  (void)in_sizes; (void)n_in; (void)out_size; (void)d_ws; (void)ws_size;
  const float* a = (const float*)d_in[0];
  const float* b = (const float*)d_in[1];
  // d_in[2] = max_displacement (known 40, baked into the tiling).
  float* out = (float*)d_out;
  dim3 grid(kW / WBLK, kH, kB);   // (4, 192, 8)
  corr_wmma_f32<<<grid, dim3(128, 1, 1), 0, stream>>>(a, b, out);
}